// ENIGMA_67302137528932
// MI455X (gfx1250) — compile-verified
//
#include <hip/hip_runtime.h>
#include <math.h>

// ---------------------------------------------------------------------------
// Problem constants (from reference)
// ---------------------------------------------------------------------------
#define BB   512
#define CC   63
#define TT   250
#define SS   10
#define EE   4
#define HIDN 184
#define PROJ 1024
#define NF   40          // conv channels
#define WOUT 46          // pooled width
#define K2   (NF*CC)     // 2520, conv2 reduction dim
#define M2   (BB*WOUT)   // 23552, conv2 rows
#define EPSV 1e-5f

typedef __attribute__((ext_vector_type(2))) float v2f;
typedef __attribute__((ext_vector_type(8))) float v8f;

#define WMMA_F32(A_, B_, C_) \
    __builtin_amdgcn_wmma_f32_16x16x4_f32(false, (A_), false, (B_), (short)0, (C_), false, false)

// ---------------------------------------------------------------------------
// Generic fp32 WMMA GEMM:  C[m,n] = sum_k A[m,k]*Bt[n,k] (+ bias[n])
// Row-major A (lda), row-major Bt (ldb) i.e. B-transposed storage.
// Batched via blockIdx.z: A += z*strideAb; Bt/bias use subj[z] if subj!=null.
// WG = 256 threads = 8 waves; tile 64(M) x 64(N); each wave: 2 16x16 tiles.
// Branchless inner loop with OOB rows clamped + float masks; wave-uniform
// fast path when the whole footprint is in-bounds.  Manual 2-stage register
// double-buffer: consume stage s while issuing loads for stage s+2, so
// s_wait_loadcnt lands behind 2 WMMAs + a full stage of independent loads.
// Requires: lda/ldb even, base pointers 8B-aligned (true for all our calls).
// ---------------------------------------------------------------------------
__global__ void gemm_wmma_f32(const float* __restrict__ A, long lda, long strideAb,
                              const float* __restrict__ Bt, long ldb, long strideBb,
                              const float* __restrict__ bias, long strideBiasB,
                              const int* __restrict__ subj,
                              float* __restrict__ C, long ldc, long strideCb,
                              int M, int N, int K)
{
    const int bz   = blockIdx.z;
    const int bsel = subj ? subj[bz] : bz;
    A    += (long)bz   * strideAb;
    Bt   += (long)bsel * strideBb;
    C    += (long)bz   * strideCb;
    const float* biasp = bias ? bias + (long)bsel * strideBiasB : nullptr;

    const int lane   = threadIdx.x & 31;
    const int wave   = threadIdx.x >> 5;   // 0..7
    const int m_sub  = wave & 3;           // 4 M-subtiles
    const int n_half = wave >> 2;          // 0..1
    const int m0 = blockIdx.y * 64 + m_sub * 16;
    const int n0 = blockIdx.x * 64 + n_half * 32;   // this wave: tiles n0, n0+16

    const int row = lane & 15;             // M (for A) / N (for B) within tile
    const int kp  = (lane >> 4) * 2;       // K offset pair: 0 or 2

    const int am   = m0 + row;
    const int bn0  = n0 + row;
    const int bn1r = n0 + 16 + row;
    const float amask  = (am   < M) ? 1.0f : 0.0f;
    const float bmask0 = (bn0  < N) ? 1.0f : 0.0f;
    const float bmask1 = (bn1r < N) ? 1.0f : 0.0f;
    // Clamp OOB rows to row 0 (always valid) -> loads are unconditional.
    const long amc  = (am   < M) ? am   : 0;
    const long bn0c = (bn0  < N) ? bn0  : 0;
    const long bn1c = (bn1r < N) ? bn1r : 0;

    const float* __restrict__ Arow  = A  + amc  * lda;
    const float* __restrict__ Brow0 = Bt + bn0c * ldb;
    const float* __restrict__ Brow1 = Bt + bn1c * ldb;

    v8f acc0 = {}; v8f acc1 = {};
    const int Ka = K & ~3;
    const int nk = Ka >> 2;                // number of K stages (4 deep each)
    // Wave-uniform: entire 16x32 footprint in bounds -> no masks needed.
    const bool full = (m0 + 16 <= M) && (n0 + 32 <= N);

    // 2-stage pipeline registers
    v2f a0 = {}, b00 = {}, b10 = {};
    v2f a1 = {}, b01 = {}, b11 = {};
    if (nk > 0) {
        a0  = *(const v2f*)(Arow  + kp);
        b00 = *(const v2f*)(Brow0 + kp);
        b10 = *(const v2f*)(Brow1 + kp);
    }
    if (nk > 1) {
        a1  = *(const v2f*)(Arow  + 4 + kp);
        b01 = *(const v2f*)(Brow0 + 4 + kp);
        b11 = *(const v2f*)(Brow1 + 4 + kp);
    }

    if (full) {
        int s = 0;
        for (; s + 4 <= nk; s += 2) {
            acc0 = WMMA_F32(a0, b00, acc0);
            acc1 = WMMA_F32(a0, b10, acc1);
            {
                const int kA = (s + 2) * 4 + kp;
                a0  = *(const v2f*)(Arow  + kA);
                b00 = *(const v2f*)(Brow0 + kA);
                b10 = *(const v2f*)(Brow1 + kA);
            }
            acc0 = WMMA_F32(a1, b01, acc0);
            acc1 = WMMA_F32(a1, b11, acc1);
            {
                const int kB = (s + 3) * 4 + kp;
                a1  = *(const v2f*)(Arow  + kB);
                b01 = *(const v2f*)(Brow0 + kB);
                b11 = *(const v2f*)(Brow1 + kB);
            }
        }
        const int rem = nk - s;            // 0..3, wave-uniform
        if (rem >= 1) { acc0 = WMMA_F32(a0, b00, acc0); acc1 = WMMA_F32(a0, b10, acc1); }
        if (rem == 3) {
            const int kA = (s + 2) * 4 + kp;
            a0  = *(const v2f*)(Arow  + kA);
            b00 = *(const v2f*)(Brow0 + kA);
            b10 = *(const v2f*)(Brow1 + kA);
        }
        if (rem >= 2) { acc0 = WMMA_F32(a1, b01, acc0); acc1 = WMMA_F32(a1, b11, acc1); }
        if (rem == 3) { acc0 = WMMA_F32(a0, b00, acc0); acc1 = WMMA_F32(a0, b10, acc1); }
    } else {
        // Same pipeline; masks applied at consumption on register copies.
#define CONSUME_MASKED(av, b0v, b1v)                                            \
        do {                                                                    \
            v2f ta = (av), tb0 = (b0v), tb1 = (b1v);                            \
            ta.x  *= amask;  ta.y  *= amask;                                    \
            tb0.x *= bmask0; tb0.y *= bmask0;                                   \
            tb1.x *= bmask1; tb1.y *= bmask1;                                   \
            acc0 = WMMA_F32(ta, tb0, acc0);                                     \
            acc1 = WMMA_F32(ta, tb1, acc1);                                     \
        } while (0)
        int s = 0;
        for (; s + 4 <= nk; s += 2) {
            CONSUME_MASKED(a0, b00, b10);
            {
                const int kA = (s + 2) * 4 + kp;
                a0  = *(const v2f*)(Arow  + kA);
                b00 = *(const v2f*)(Brow0 + kA);
                b10 = *(const v2f*)(Brow1 + kA);
            }
            CONSUME_MASKED(a1, b01, b11);
            {
                const int kB = (s + 3) * 4 + kp;
                a1  = *(const v2f*)(Arow  + kB);
                b01 = *(const v2f*)(Brow0 + kB);
                b11 = *(const v2f*)(Brow1 + kB);
            }
        }
        const int rem = nk - s;
        if (rem >= 1) CONSUME_MASKED(a0, b00, b10);
        if (rem == 3) {
            const int kA = (s + 2) * 4 + kp;
            a0  = *(const v2f*)(Arow  + kA);
            b00 = *(const v2f*)(Brow0 + kA);
            b10 = *(const v2f*)(Brow1 + kA);
        }
        if (rem >= 2) CONSUME_MASKED(a1, b01, b11);
        if (rem == 3) CONSUME_MASKED(a0, b00, b10);
#undef CONSUME_MASKED
    }

    if (Ka < K) {   // K-tail (rem 1..3 elements): one masked step, clamped indices
        const int k = Ka + kp;
        const float mk0 = (k     < K) ? 1.0f : 0.0f;
        const float mk1 = (k + 1 < K) ? 1.0f : 0.0f;
        const int kc0 = (k     < K) ? k     : K - 1;
        const int kc1 = (k + 1 < K) ? k + 1 : K - 1;
        v2f a, b0, b1;
        a.x  = Arow [kc0] * mk0 * amask;   a.y  = Arow [kc1] * mk1 * amask;
        b0.x = Brow0[kc0] * mk0 * bmask0;  b0.y = Brow0[kc1] * mk1 * bmask0;
        b1.x = Brow1[kc0] * mk0 * bmask1;  b1.y = Brow1[kc1] * mk1 * bmask1;
        acc0 = WMMA_F32(a, b0, acc0);
        acc1 = WMMA_F32(a, b1, acc1);
    }

    // C/D layout: VGPR r -> lanes 0-15: M=r, N=lane ; lanes 16-31: M=8+r, N=lane-16
    const int n_st0  = n0 + (lane & 15);
    const int n_st1  = n0 + 16 + (lane & 15);
    const int m_base = m0 + ((lane >> 4) * 8);
    const float bv0 = (biasp && n_st0 < N) ? biasp[n_st0] : 0.0f;
    const float bv1 = (biasp && n_st1 < N) ? biasp[n_st1] : 0.0f;
#pragma unroll
    for (int r = 0; r < 8; ++r) {
        const int m = m_base + r;
        if (m < M) {
            if (n_st0 < N) C[(long)m * ldc + n_st0] = acc0[r] + bv0;
            if (n_st1 < N) C[(long)m * ldc + n_st1] = acc1[r] + bv1;
        }
    }
}

// ---------------------------------------------------------------------------
// Small helper kernels
// ---------------------------------------------------------------------------
__global__ void k_zero(float* __restrict__ p, int n)
{
    int i = blockIdx.x * blockDim.x + threadIdx.x;
    if (i < n) p[i] = 0.0f;
}

// Sliding window-17 sums along T:  sx[b,c,t0] = sum_{i<17} xe[b,c,t0+i], t0<230
__global__ void k_winsum17(const float* __restrict__ xe, float* __restrict__ sx)
{
    const long total = (long)BB * CC * 230;
    long i = (long)blockIdx.x * blockDim.x + threadIdx.x;
    if (i >= total) return;
    long rc = i / 230;
    int  t0 = (int)(i % 230);
    const float* row = xe + rc * (long)TT + t0;
    float s = 0.0f;
#pragma unroll
    for (int j = 0; j < 17; ++j) s += row[j];
    sx[i] = s;
}

// Fused conv1(1x5) + avgpool(17,5):  P[b,u,f,c] = b1[f] + (1/17)*sum_j w1[f,j]*sx[b,c,5u+j]
// P stored contiguously as (B, WOUT, NF, CC) so conv2 becomes a plain GEMM.
__global__ void k_conv1pool(const float* __restrict__ sx, const float* __restrict__ w1,
                            const float* __restrict__ b1, float* __restrict__ P)
{
    const long total = (long)BB * WOUT * NF * CC;
    long i = (long)blockIdx.x * blockDim.x + threadIdx.x;
    if (i >= total) return;
    int  c = (int)(i % CC);   long r = i / CC;
    int  f = (int)(r % NF);   r /= NF;
    int  u = (int)(r % WOUT); long b = r / WOUT;
    const float* s = sx + (b * CC + c) * 230L + 5 * u;
    float acc = 0.0f;
#pragma unroll
    for (int j = 0; j < 5; ++j) acc += w1[f * 5 + j] * s[j];
    P[i] = acc * (1.0f / 17.0f) + b1[f];
}

// Per-channel sum / sum-of-squares.  channel = (i / inner) % nchan.
// LDS-accumulated partials, one global atomic per channel per block.
__global__ void k_chan_stats(const float* __restrict__ X, long n, int nchan, long inner,
                             float* __restrict__ sums)
{
    __shared__ float ls[128];   // supports nchan <= 64
    for (int i = threadIdx.x; i < 2 * nchan; i += blockDim.x) ls[i] = 0.0f;
    __syncthreads();
    const long stride = (long)gridDim.x * blockDim.x;
    for (long i = (long)blockIdx.x * blockDim.x + threadIdx.x; i < n; i += stride) {
        float v  = X[i];
        int   ch = (int)((i / inner) % nchan);
        atomicAdd(&ls[ch], v);
        atomicAdd(&ls[nchan + ch], v * v);
    }
    __syncthreads();
    for (int i = threadIdx.x; i < 2 * nchan; i += blockDim.x)
        unsafeAtomicAdd(&sums[i], ls[i]);
}

// Turn (sum, sumsq) into BN scale/shift: ss[f]=g*rsqrt(var+eps), ss[nc+f]=b-mu*scale
__global__ void k_bn_finalize(const float* __restrict__ sums, float count,
                              const float* __restrict__ g, const float* __restrict__ b,
                              float* __restrict__ ss, int nchan)
{
    int f = blockIdx.x * blockDim.x + threadIdx.x;
    if (f >= nchan) return;
    float mu  = sums[f] / count;
    float var = sums[nchan + f] / count - mu * mu;
    float sc  = g[f] * rsqrtf(var + EPSV);
    ss[f]         = sc;
    ss[nchan + f] = b[f] - mu * sc;
}

// In-place BN + ELU:  X = elu(X*scale[ch] + shift[ch])
__global__ void k_bn_elu(float* __restrict__ X, long n, int nchan, long inner,
                         const float* __restrict__ ss)
{
    long i = (long)blockIdx.x * blockDim.x + threadIdx.x;
    if (i >= n) return;
    int ch  = (int)((i / inner) % nchan);
    float v = X[i] * ss[ch] + ss[nchan + ch];
    X[i] = v > 0.0f ? v : expm1f(v);
}

// BN2 + ELU + 1x1 projection + NHWC flatten:
// z[b, u*4+e] = pb[e] + sum_o pw[e,o]*elu(C2[(b*46+u),o]*s2[o]+t2[o])
__global__ void k_bn2_elu_proj(const float* __restrict__ C2, const float* __restrict__ ss,
                               const float* __restrict__ pw, const float* __restrict__ pb,
                               float* __restrict__ z)
{
    int i = blockIdx.x * blockDim.x + threadIdx.x;   // (b*46+u)
    if (i >= M2) return;
    const float* row = C2 + (long)i * NF;
    float acc0 = pb[0], acc1 = pb[1], acc2 = pb[2], acc3 = pb[3];
    for (int o = 0; o < NF; ++o) {
        float v = row[o] * ss[o] + ss[NF + o];
        v = v > 0.0f ? v : expm1f(v);
        acc0 += pw[0 * NF + o] * v;
        acc1 += pw[1 * NF + o] * v;
        acc2 += pw[2 * NF + o] * v;
        acc3 += pw[3 * NF + o] * v;
    }
    int b = i / WOUT, u = i % WOUT;
    float* zp = z + (long)b * HIDN + u * EE;
    zp[0] = acc0; zp[1] = acc1; zp[2] = acc2; zp[3] = acc3;
}

// Exact GELU: 0.5*x*(1+erf(x/sqrt(2)))
__global__ void k_gelu(const float* __restrict__ h1, float* __restrict__ gg, long n)
{
    long i = (long)blockIdx.x * blockDim.x + threadIdx.x;
    if (i >= n) return;
    float x = h1[i];
    gg[i] = 0.5f * x * (1.0f + erff(x * 0.70710678118654752f));
}

// Residual add + LayerNorm over last dim (1024).  One 256-thread block per row.
__global__ void k_res_ln(const float* __restrict__ h1, const float* __restrict__ r,
                         const float* __restrict__ g, const float* __restrict__ b,
                         float* __restrict__ out)
{
    __shared__ float s1[256];
    __shared__ float s2[256];
    const int row = blockIdx.x, t = threadIdx.x;
    const float* A = h1 + (long)row * PROJ;
    const float* R = r  + (long)row * PROJ;
    float v[4]; float sm = 0.0f, sq = 0.0f;
#pragma unroll
    for (int j = 0; j < 4; ++j) {
        v[j] = A[t + 256 * j] + R[t + 256 * j];
        sm += v[j]; sq += v[j] * v[j];
    }
    s1[t] = sm; s2[t] = sq;
    __syncthreads();
    for (int off = 128; off > 0; off >>= 1) {
        if (t < off) { s1[t] += s1[t + off]; s2[t] += s2[t + off]; }
        __syncthreads();
    }
    const float mu  = s1[0] * (1.0f / PROJ);
    const float var = s2[0] * (1.0f / PROJ) - mu * mu;
    const float inv = rsqrtf(var + EPSV);
#pragma unroll
    for (int j = 0; j < 4; ++j) {
        int col = t + 256 * j;
        out[(long)row * PROJ + col] = (v[j] - mu) * inv * g[col] + b[col];
    }
}

// ---------------------------------------------------------------------------
// Launcher
// ---------------------------------------------------------------------------
extern "C" void kernel_launch(void* const* d_in, const int* in_sizes, int n_in,
                              void* d_out, int out_size, void* d_ws, size_t ws_size,
                              hipStream_t stream)
{
    const float* x        = (const float*)d_in[0];
    const int*   subjects = (const int*)  d_in[1];
    const float* W_subj   = (const float*)d_in[2];
    const float* b_subj   = (const float*)d_in[3];
    const float* conv1_w  = (const float*)d_in[4];
    const float* conv1_b  = (const float*)d_in[5];
    const float* bn1_g    = (const float*)d_in[6];
    const float* bn1_b    = (const float*)d_in[7];
    const float* conv2_w  = (const float*)d_in[8];
    const float* conv2_b  = (const float*)d_in[9];
    const float* bn2_g    = (const float*)d_in[10];
    const float* bn2_b    = (const float*)d_in[11];
    const float* proj_w   = (const float*)d_in[12];
    const float* proj_b   = (const float*)d_in[13];
    const float* fc1_w    = (const float*)d_in[14];
    const float* fc1_b    = (const float*)d_in[15];
    const float* fc2_w    = (const float*)d_in[16];
    const float* fc2_b    = (const float*)d_in[17];
    const float* ln_g     = (const float*)d_in[18];
    const float* ln_b     = (const float*)d_in[19];
    float* out = (float*)d_out;

    // Workspace layout (floats). Total ~77.5M floats (~296 MB).
    float* ws    = (float*)d_ws;
    float* stats = ws;                               // 1024
    float* xeeg  = ws    + 1024;                     // 512*63*250   = 8,064,000
    float* sx    = xeeg  + (long)BB * CC * TT;       // 512*63*230   = 7,418,880
    float* P     = sx    + (long)BB * CC * 230;      // 512*46*40*63 = 59,351,040
    float* C2    = P     + (long)BB * WOUT * NF * CC;// 23552*40     = 942,080
    float* z     = C2    + (long)M2 * NF;            // 512*184      = 94,208
    float* h1    = z     + (long)BB * HIDN;          // 512*1024
    float* gg    = h1    + (long)BB * PROJ;          // 512*1024
    float* rr    = gg    + (long)BB * PROJ;          // 512*1024

    float* bn1_sums = stats;        // [0..79]
    float* bn1_ss   = stats + 128;  // scale[40], shift[40]
    float* bn2_sums = stats + 256;
    float* bn2_ss   = stats + 384;

    k_zero<<<4, 256, 0, stream>>>(stats, 1024);

    // 1) Subject-specific spatial transform (batched GEMM, gather via subjects):
    //    xeeg[b] (63x250) = x[b] (63x250) @ W_subj[subjects[b]]^T + b_subj[subjects[b]]
    {
        dim3 g((TT + 63) / 64, (CC + 63) / 64, BB);
        gemm_wmma_f32<<<g, 256, 0, stream>>>(
            x, TT, (long)CC * TT,
            W_subj, TT, (long)TT * TT,
            b_subj, TT, subjects,
            xeeg, TT, (long)CC * TT,
            CC, TT, TT);
    }

    // 2) Window-17 sliding sums (factors conv1+avgpool: 10 GF recompute -> ~0.3 GF)
    {
        long n = (long)BB * CC * 230;
        k_winsum17<<<(unsigned)((n + 255) / 256), 256, 0, stream>>>(xeeg, sx);
    }

    // 3) conv1 + avgpool fused, written in (b,u,f,c) layout for conv2-as-GEMM
    {
        long n = (long)BB * WOUT * NF * CC;
        k_conv1pool<<<(unsigned)((n + 255) / 256), 256, 0, stream>>>(sx, conv1_w, conv1_b, P);
    }

    // 4) BN1 stats + finalize + in-place BN+ELU
    {
        long n = (long)BB * WOUT * NF * CC;
        k_chan_stats<<<2048, 256, 0, stream>>>(P, n, NF, CC, bn1_sums);
        k_bn_finalize<<<1, 64, 0, stream>>>(bn1_sums, (float)((long)BB * WOUT * CC),
                                            bn1_g, bn1_b, bn1_ss, NF);
        k_bn_elu<<<(unsigned)((n + 255) / 256), 256, 0, stream>>>(P, n, NF, CC, bn1_ss);
    }

    // 5) conv2 as GEMM: C2 (23552x40) = P (23552x2520) @ conv2_w^T + conv2_b
    //    M = 368*64 exactly -> all M-tiles full; only n_half=1 waves masked.
    {
        dim3 g((NF + 63) / 64, (M2 + 63) / 64, 1);
        gemm_wmma_f32<<<g, 256, 0, stream>>>(
            P, K2, 0, conv2_w, K2, 0, conv2_b, 0, nullptr,
            C2, NF, 0, M2, NF, K2);
    }

    // 6) BN2 stats + finalize, then fused BN2+ELU+proj(1x1)+flatten -> z (512x184)
    {
        long n = (long)M2 * NF;
        k_chan_stats<<<512, 256, 0, stream>>>(C2, n, NF, 1, bn2_sums);
        k_bn_finalize<<<1, 64, 0, stream>>>(bn2_sums, (float)M2, bn2_g, bn2_b, bn2_ss, NF);
        k_bn2_elu_proj<<<(M2 + 255) / 256, 256, 0, stream>>>(C2, bn2_ss, proj_w, proj_b, z);
    }

    // 7) fc1: h1 (512x1024) = z (512x184) @ fc1_w^T + fc1_b   (fully aligned)
    {
        dim3 g((PROJ + 63) / 64, (BB + 63) / 64, 1);
        gemm_wmma_f32<<<g, 256, 0, stream>>>(
            z, HIDN, 0, fc1_w, HIDN, 0, fc1_b, 0, nullptr,
            h1, PROJ, 0, BB, PROJ, HIDN);
    }

    // 8) exact GELU
    {
        long n = (long)BB * PROJ;
        k_gelu<<<(unsigned)((n + 255) / 256), 256, 0, stream>>>(h1, gg, n);
    }

    // 9) fc2: rr = gelu(h1) @ fc2_w^T + fc2_b   (fully aligned)
    {
        dim3 g((PROJ + 63) / 64, (BB + 63) / 64, 1);
        gemm_wmma_f32<<<g, 256, 0, stream>>>(
            gg, PROJ, 0, fc2_w, PROJ, 0, fc2_b, 0, nullptr,
            rr, PROJ, 0, BB, PROJ, PROJ);
    }

    // 10) h2 = h1 + rr ; LayerNorm -> out
    k_res_ln<<<BB, 256, 0, stream>>>(h1, rr, ln_g, ln_b, out);
}